// Upsample_62534723829812
// MI455X (gfx1250) — compile-verified
//
#include <hip/hip_runtime.h>

// Pure bandwidth kernel: out[:, :, ::2, ::2] = x, zeros elsewhere.
// 128 MiB read + 512 MiB written, 0 FLOPs -> HBM-roofline ~28.8 us on MI455X.
// Even output rows: b128 NT stores of value/zero interleaved vectors.
// Odd (all-zero) output rows: gfx1250 async LDS->global b128 stores (ASYNCcnt).

typedef float v4f __attribute__((ext_vector_type(4)));
typedef int v4i __attribute__((vector_size(4 * sizeof(int))));  // matches builtin param

#ifndef __has_builtin
#define __has_builtin(x) 0
#endif

#if __has_builtin(__builtin_amdgcn_global_store_async_from_lds_b128)
#define USE_ASYNC_ZERO 1
#else
#define USE_ASYNC_ZERO 0
#endif

// Fixed problem geometry from the reference: x is (32, 256, 64, 64) fp32.
#define IN_W 64u           // input width (floats)
#define IN_H 64u           // input height
#define OUT_W 128u         // output width (floats)
#define WGROUPS 16u        // IN_W / 4 float4-groups per input row

#if USE_ASYNC_ZERO
typedef __attribute__((address_space(1))) v4i* gv4i_ptr;  // global
typedef __attribute__((address_space(3))) v4i* lv4i_ptr;  // LDS
#endif

__global__ __launch_bounds__(256) void upsample_scatter_kernel(
    const float* __restrict__ x, float* __restrict__ out, unsigned ngroups) {
#if USE_ASYNC_ZERO
  // Per-thread 16B zero slot in LDS plus one 16B pad slot: if the compiler
  // folds the second async store as inst_offset 16, the ISA adds that offset
  // to the LDS address too, and lane 255 then reads the pad (still zero,
  // still in-bounds).
  __shared__ __align__(16) float zlds[256 * 4 + 4];
  {
    v4f z = {0.f, 0.f, 0.f, 0.f};
    *(v4f*)&zlds[threadIdx.x * 4u] = z;
    if (threadIdx.x == 0) *(v4f*)&zlds[256 * 4] = z;
  }
  __syncthreads();  // s_wait_dscnt 0 + barrier: LDS zeros visible before async reads
#endif

  unsigned gid = blockIdx.x * 256u + threadIdx.x;
  if (gid >= ngroups) return;

  unsigned wg    = gid & (WGROUPS - 1u);   // which float4 within the input row
  unsigned rowIn = gid >> 4;               // global input row index = bc*IN_H + h
  unsigned h     = rowIn & (IN_H - 1u);
  unsigned bc    = rowIn >> 6;             // fused batch*channel index

  // 16B b128 non-temporal load of 4 consecutive input floats (read-once data).
  const v4f* src = (const v4f*)(x + rowIn * IN_W + wg * 4u);
  v4f v = __builtin_nontemporal_load(src);

  // Even output row: interleave zeros so every store is a full aligned b128.
  unsigned evenBase = (bc * (2u * IN_H) + 2u * h) * OUT_W + wg * 8u;
  v4f lo = {v.x, 0.f, v.y, 0.f};
  v4f hi = {v.z, 0.f, v.w, 0.f};
  __builtin_nontemporal_store(lo, (v4f*)(out + evenBase));
  __builtin_nontemporal_store(hi, (v4f*)(out + evenBase + 4u));

  // Odd output row: all zeros, streamed from LDS via async b128 stores.
  unsigned oddBase = evenBase + OUT_W;
#if USE_ASYNC_ZERO
  {
    lv4i_ptr lsrc = (lv4i_ptr)&zlds[threadIdx.x * 4u];
    gv4i_ptr g0 = (gv4i_ptr)(out + oddBase);
    gv4i_ptr g1 = (gv4i_ptr)(out + oddBase + 4u);
    // global_store_async_from_lds_b128 (ASYNCcnt); implicit S_WAIT_IDLE at
    // S_ENDPGM guarantees completion before LDS/wave teardown.
    __builtin_amdgcn_global_store_async_from_lds_b128(g0, lsrc, 0, 0);
    __builtin_amdgcn_global_store_async_from_lds_b128(g1, lsrc, 0, 0);
  }
#else
  {
    v4f z = {0.f, 0.f, 0.f, 0.f};
    __builtin_nontemporal_store(z, (v4f*)(out + oddBase));
    __builtin_nontemporal_store(z, (v4f*)(out + oddBase + 4u));
  }
#endif
}

extern "C" void kernel_launch(void* const* d_in, const int* in_sizes, int n_in,
                              void* d_out, int out_size, void* d_ws, size_t ws_size,
                              hipStream_t stream) {
  (void)n_in; (void)out_size; (void)d_ws; (void)ws_size;
  const float* x = (const float*)d_in[0];
  float* out = (float*)d_out;

  unsigned n = (unsigned)in_sizes[0];      // 32*256*64*64 = 33,554,432
  unsigned ngroups = n / 4u;               // one float4 per thread = 8,388,608
  unsigned blocks = (ngroups + 255u) / 256u;  // 32768 blocks of 8 wave32s

  upsample_scatter_kernel<<<blocks, 256, 0, stream>>>(x, out, ngroups);
}